// GCN_22093311771169
// MI455X (gfx1250) — compile-verified
//
#include <hip/hip_runtime.h>
#include <hip/hip_bf16.h>

typedef float v2f __attribute__((ext_vector_type(2)));
typedef float v8f __attribute__((ext_vector_type(8)));

// ---------------------------------------------------------------------------
// Degree / normalization: deg starts at 1 (self loop), add 1 per in-edge,
// then dinv = rsqrt(deg). norm(s,d) = dinv[s]*dinv[d].
// ---------------------------------------------------------------------------
__global__ void deg_init_k(float* __restrict__ deg, int n) {
    int i = blockIdx.x * blockDim.x + threadIdx.x;
    if (i < n) deg[i] = 1.0f;
}

__global__ void deg_accum_k(const int* __restrict__ dst, float* __restrict__ deg, int e) {
    int i = blockIdx.x * blockDim.x + threadIdx.x;
    if (i < e) atomicAdd(&deg[dst[i]], 1.0f);
}

__global__ void deg_rsqrt_k(float* __restrict__ deg, int n) {
    int i = blockIdx.x * blockDim.x + threadIdx.x;
    if (i < n) deg[i] = rsqrtf(deg[i]);
}

// ---------------------------------------------------------------------------
// FP32 WMMA GEMM: C[M x N] = op(A)[M x K] @ B[K x N]
// op(A) = RELU ? max(A + bias, 0) : A   (fuses GCN bias+ReLU into layer-2 load)
// One wave computes one 16x16 tile using V_WMMA_F32_16X16X4_F32, K/4 steps.
// A frag (16x4): lane<16 -> row=lane&15, k = k0+{0,1}; lane>=16 -> k = k0+{2,3}
// B frag (4x16): lane<16 -> col=lane&15, k = k0+{0,1}; lane>=16 -> k = k0+{2,3}
// C/D (16x16):   vgpr r -> row = 16*mT + r + (lane>=16 ? 8:0), col = lane&15
//
// M-boundary handling: CLAMP the A row instead of predicating the load.
// Out-of-range lanes read row M-1; each WMMA output row depends only on its
// own A row, so garbage lands only in rows the guarded store discards.
// This keeps the K-loop branch-free (no exec save/restore), so loads batch
// into s_clause groups instead of serializing each WMMA on s_wait_loadcnt 0.
// Loads are issued 8 K-steps ahead of the WMMA chain (chunked double-phase).
// ---------------------------------------------------------------------------
template <int K, int N, int RELU>
__global__ void wmma_gemm_k(const float* __restrict__ A, const float* __restrict__ B,
                            const float* __restrict__ bias, float* __restrict__ C, int M) {
    const int lane   = threadIdx.x & 31;
    const int wave   = threadIdx.x >> 5;
    const int nTiles = N / 16;
    const int mTiles = (M + 15) >> 4;
    const int tile   = blockIdx.x * (blockDim.x >> 5) + wave;
    if (tile >= mTiles * nTiles) return;          // wave-uniform exit

    const int mT   = tile / nTiles;
    const int nT   = tile % nTiles;
    const int rowA = mT * 16 + (lane & 15);
    const int col  = nT * 16 + (lane & 15);
    const int kOff = (lane >> 4) << 1;            // 0 for lanes 0-15, 2 for 16-31
    const int rowAc = (rowA < M) ? rowA : (M - 1);   // clamp, not predicate

    const float* __restrict__ Arow = A + (size_t)rowAc * K;
    const float* __restrict__ Bcol = B + col;

    constexpr int STEPS = K / 4;                  // 16 (K=64) or 32 (K=128)
    constexpr int CH    = 8;                      // K-steps per load/compute chunk

    v8f acc = {0.f, 0.f, 0.f, 0.f, 0.f, 0.f, 0.f, 0.f};
#pragma unroll
    for (int c = 0; c < STEPS; c += CH) {
        v2f af[CH], bf[CH];
        // phase 1: issue all 3*CH loads back-to-back (clause-friendly)
#pragma unroll
        for (int j = 0; j < CH; ++j) {
            const int ka = (c + j) * 4 + kOff;
            float a0 = Arow[ka];
            float a1 = Arow[ka + 1];
            if (RELU) {
                a0 = fmaxf(a0 + bias[ka], 0.0f);
                a1 = fmaxf(a1 + bias[ka + 1], 0.0f);
            }
            af[j].x = a0; af[j].y = a1;
            bf[j].x = Bcol[(size_t)ka * N];
            bf[j].y = Bcol[(size_t)(ka + 1) * N];
        }
        // phase 2: WMMA chain
#pragma unroll
        for (int j = 0; j < CH; ++j) {
            // 8-arg form: (neg_a, A, neg_b, B, c_mod, C, reuse_a, reuse_b)
            acc = __builtin_amdgcn_wmma_f32_16x16x4_f32(false, af[j], false, bf[j],
                                                        (short)0, acc, false, false);
        }
    }

    const int rowBase = mT * 16 + ((lane >> 4) << 3);   // +8 for upper lane half
#pragma unroll
    for (int r = 0; r < 8; ++r) {
        const int row = rowBase + r;
        if (row < M) C[(size_t)row * N + col] = acc[r];
    }
}

// ---------------------------------------------------------------------------
// Aggregation. Self-loop term initializes agg (no atomics): agg[i] = h[i]*dinv[i]^2.
// Then one thread per (edge, 4-feature chunk): coalesced float4 gather from h
// (L2-resident: 25.6 MB << 192 MB L2) + 4x global_atomic_add_f32 scatter.
// ---------------------------------------------------------------------------
template <int F>
__global__ void agg_init_k(const float* __restrict__ h, const float* __restrict__ dinv,
                           float* __restrict__ agg, int n) {
    long long idx = (long long)blockIdx.x * blockDim.x + threadIdx.x;
    if (idx >= (long long)n * F) return;
    int i = (int)(idx / F);
    float w = dinv[i] * dinv[i];
    agg[idx] = h[idx] * w;
}

template <int F>
__global__ void agg_edges_k(const int* __restrict__ src, const int* __restrict__ dst,
                            const float* __restrict__ dinv, const float* __restrict__ h,
                            float* __restrict__ agg, int E) {
    const int chunks = F >> 2;
    long long t = (long long)blockIdx.x * blockDim.x + threadIdx.x;
    int e = (int)(t / chunks);
    if (e >= E) return;
    int c = ((int)(t % chunks)) << 2;
    int s = src[e], d = dst[e];
    float w = dinv[s] * dinv[d];
    const float4 hv = *(const float4*)(h + (size_t)s * F + c);
    float* out = agg + (size_t)d * F + c;
    atomicAdd(out + 0, hv.x * w);
    atomicAdd(out + 1, hv.y * w);
    atomicAdd(out + 2, hv.z * w);
    atomicAdd(out + 3, hv.w * w);
}

// ---------------------------------------------------------------------------
// Global mean pool over graphs (+ final bias b2 folded into the sum).
// ---------------------------------------------------------------------------
__global__ void zero_k(float* __restrict__ p, int n) {
    int i = blockIdx.x * blockDim.x + threadIdx.x;
    if (i < n) p[i] = 0.0f;
}

__global__ void count_k(const int* __restrict__ batch, float* __restrict__ counts, int n) {
    int i = blockIdx.x * blockDim.x + threadIdx.x;
    if (i < n) atomicAdd(&counts[batch[i]], 1.0f);
}

__global__ void pool_sum_k(const float* __restrict__ agg, const float* __restrict__ b2,
                           const int* __restrict__ batch, float* __restrict__ out, int n) {
    long long idx = (long long)blockIdx.x * blockDim.x + threadIdx.x;
    if (idx >= (long long)n * 64) return;
    int i = (int)(idx >> 6);
    int f = (int)(idx & 63);
    atomicAdd(&out[(size_t)batch[i] * 64 + f], agg[idx] + b2[f]);
}

__global__ void pool_div_k(float* __restrict__ out, const float* __restrict__ counts, int g) {
    int idx = blockIdx.x * blockDim.x + threadIdx.x;
    if (idx >= g * 64) return;
    out[idx] /= fmaxf(counts[idx >> 6], 1.0f);
}

// ---------------------------------------------------------------------------
static inline int cdivll(long long a, long long b) { return (int)((a + b - 1) / b); }

extern "C" void kernel_launch(void* const* d_in, const int* in_sizes, int n_in,
                              void* d_out, int out_size, void* d_ws, size_t ws_size,
                              hipStream_t stream) {
    const float* x     = (const float*)d_in[0];
    const int*   ei    = (const int*)d_in[1];     // [2, E]: row0=src, row1=dst
    const int*   batch = (const int*)d_in[2];
    const float* W1    = (const float*)d_in[3];   // [64,128]
    const float* b1    = (const float*)d_in[4];   // [128]
    const float* W2    = (const float*)d_in[5];   // [128,64]
    const float* b2    = (const float*)d_in[6];   // [64]
    float*       out   = (float*)d_out;

    const int Nn = in_sizes[0] / 64;   // 50000 nodes
    const int E  = in_sizes[1] / 2;    // 800000 edges
    const int G  = out_size / 64;      // 512 graphs
    const int* src = ei;
    const int* dst = ei + E;

    // Workspace layout (floats), buffers aliased across the two layers:
    //   dinv  [Nn]            (deg, then rsqrt in place)
    //   bufA  [Nn*128]        h1, later h2 (first Nn*64)
    //   bufB  [Nn*128]        agg1, later agg2 (first Nn*64)
    //   counts[G]
    float* ws     = (float*)d_ws;
    float* dinv   = ws;
    float* bufA   = dinv + (((size_t)Nn + 255) & ~(size_t)255);
    float* bufB   = bufA + (size_t)Nn * 128;
    float* counts = bufB + (size_t)Nn * 128;

    const int T = 256;

    // --- degree / symmetric normalization ---
    deg_init_k<<<cdivll(Nn, T), T, 0, stream>>>(dinv, Nn);
    deg_accum_k<<<cdivll(E, T), T, 0, stream>>>(dst, dinv, E);
    deg_rsqrt_k<<<cdivll(Nn, T), T, 0, stream>>>(dinv, Nn);

    const int mTiles = (Nn + 15) / 16;

    // --- layer 1: h1 = x @ W1 (fp32 WMMA), then normalized scatter ---
    wmma_gemm_k<64, 128, 0><<<cdivll((long long)mTiles * 8, 8), T, 0, stream>>>(
        x, W1, nullptr, bufA, Nn);
    agg_init_k<128><<<cdivll((long long)Nn * 128, T), T, 0, stream>>>(bufA, dinv, bufB, Nn);
    agg_edges_k<128><<<cdivll((long long)E * 32, T), T, 0, stream>>>(src, dst, dinv, bufA, bufB, E);

    // --- layer 2: h2 = relu(agg1 + b1) @ W2 (bias+ReLU fused into A load) ---
    wmma_gemm_k<128, 64, 1><<<cdivll((long long)mTiles * 4, 8), T, 0, stream>>>(
        bufB, W2, b1, bufA, Nn);
    agg_init_k<64><<<cdivll((long long)Nn * 64, T), T, 0, stream>>>(bufA, dinv, bufB, Nn);
    agg_edges_k<64><<<cdivll((long long)E * 16, T), T, 0, stream>>>(src, dst, dinv, bufA, bufB, E);

    // --- global mean pool (+b2) ---
    zero_k<<<cdivll(G * 64, T), T, 0, stream>>>(out, G * 64);
    zero_k<<<cdivll(G, T), T, 0, stream>>>(counts, G);
    count_k<<<cdivll(Nn, T), T, 0, stream>>>(batch, counts, Nn);
    pool_sum_k<<<cdivll((long long)Nn * 64, T), T, 0, stream>>>(bufB, b2, batch, out, Nn);
    pool_div_k<<<cdivll(G * 64, T), T, 0, stream>>>(out, counts, G);
}